// GraphAttentionNetwork_10909216932560
// MI455X (gfx1250) — compile-verified
//
#include <hip/hip_runtime.h>
#include <math.h>

// ---------------------------------------------------------------------------
// GAT (3-layer, PyG GATConv semantics, eval mode) for MI455X / gfx1250.
// Dense projections: native FP32 WMMA pipe (V_WMMA_F32_16X16X4_F32), one wave
// per 16-row strip, A-fragment reused across all N-tiles; per K-step all NT
// B-fragments are loaded first, then NT WMMAs issue back-to-back (single
// loadcnt wait per step instead of one per WMMA). Weights zero-padded so no
// per-lane predication (EXEC stays all-ones, loads unconditional).
// Edge phase (segment softmax + scatter) uses native float atomics; that is
// the real bandwidth bottleneck (L2-resident gathers + atomic adds).
// ---------------------------------------------------------------------------

typedef __attribute__((ext_vector_type(2))) float v2f;
typedef __attribute__((ext_vector_type(8))) float v8f;

#define NEG_SLOPE 0.2f
#define BN_EPS 1e-5f
#define KDIM 128   // inner dimension of every projection in this net

// ---------------- FP32 WMMA GEMM: Hout[M,ldW] = X[M,128] @ W[128,ldW] -------
// NT = number of 16-wide N-tiles (ldW == NT*16). One wave per 16-row strip.
// A-frag layout (ISA 7.12.2, 32-bit A 16x4): lanes 0-15 hold K={0,1},
// lanes 16-31 hold K={2,3}; M = lane&15 in both halves. B mirrors on K.
// C/D: VGPR r, lane l -> row r + 8*(l>=16), col l&15.
template <int NT>
__global__ void gat_gemm_wmma_f32(const float* __restrict__ X,
                                  const float* __restrict__ W,
                                  float* __restrict__ Hout,
                                  int M, int ldW) {
    const int wid    = (blockIdx.x * blockDim.x + threadIdx.x) >> 5;
    const int mtiles = M >> 4;              // M is a multiple of 16
    if (wid >= mtiles) return;

    const int lane = threadIdx.x & 31;
    const int half = lane >> 4;             // 0: K={0,1}, 1: K={2,3}
    const int l16  = lane & 15;

    const float* Xp = X + (long)(wid * 16 + l16) * KDIM + 2 * half;
    const float* Wp = W + (long)(2 * half) * ldW + l16;

    v8f acc[NT];
#pragma unroll
    for (int t = 0; t < NT; ++t) acc[t] = (v8f){};

#pragma unroll 2
    for (int k = 0; k < KDIM; k += 4) {
        // ---- load phase: A fragment + all NT B fragments -------------------
        v2f a;
        a.x = Xp[k];
        a.y = Xp[k + 1];
        const float* wk = Wp + (long)k * ldW;
        v2f b[NT];
#pragma unroll
        for (int t = 0; t < NT; ++t) {
            b[t].x = wk[t * 16];
            b[t].y = wk[ldW + t * 16];
        }
        // ---- compute phase: NT back-to-back WMMAs --------------------------
#pragma unroll
        for (int t = 0; t < NT; ++t) {
            acc[t] = __builtin_amdgcn_wmma_f32_16x16x4_f32(
                /*neg_a=*/false, a, /*neg_b=*/false, b[t],
                /*c_mod=*/(short)0, acc[t], /*reuse_a=*/false, /*reuse_b=*/false);
        }
    }

    float* out = Hout + (long)(wid * 16) * ldW + l16;
#pragma unroll
    for (int t = 0; t < NT; ++t) {
#pragma unroll
        for (int r = 0; r < 8; ++r) {
            out[(long)(r + 8 * half) * ldW + t * 16] = acc[t][r];
        }
    }
}

// --------------- zero-pad W[K,Nout] -> Wp[K,Npad] --------------------------
__global__ void gat_pad_weight(const float* __restrict__ W, float* __restrict__ Wp,
                               int K, int Nout, int Npad) {
    const int i = blockIdx.x * blockDim.x + threadIdx.x;
    if (i >= K * Npad) return;
    const int r = i / Npad, c = i % Npad;
    Wp[i] = (c < Nout) ? W[r * Nout + c] : 0.0f;
}

// --------------- attention coefficients per (node, head) -------------------
__global__ void gat_al_kernel(const float* __restrict__ hproj, int ldH,
                              const float* __restrict__ a_src,
                              const float* __restrict__ a_dst,
                              float* __restrict__ alS, float* __restrict__ alD,
                              int n, int H, int C) {
    const int t = blockIdx.x * blockDim.x + threadIdx.x;
    if (t >= n * H) return;
    const int node = t / H, hh = t % H;
    const float* row = hproj + (long)node * ldH + hh * C;
    float s1 = 0.0f, s2 = 0.0f;
    for (int c = 0; c < C; ++c) {
        const float v = row[c];
        s1 += v * a_src[hh * C + c];
        s2 += v * a_dst[hh * C + c];
    }
    alS[t] = s1;
    alD[t] = s2;
}

// --------------- init: out=bias, max=-inf, denom=0 -------------------------
__global__ void gat_init_kernel(float* __restrict__ aggout,
                                float* __restrict__ mmax,
                                float* __restrict__ denom,
                                const float* __restrict__ bias,
                                int n, int HC, int H) {
    const int i = blockIdx.x * blockDim.x + threadIdx.x;
    if (i < n * HC) aggout[i] = bias[i % HC];
    if (i < n * H) {
        mmax[i]  = __int_as_float(0xff800000);  // -inf
        denom[i] = 0.0f;
    }
}

// --------------- edge helpers ----------------------------------------------
__device__ __forceinline__ void edge_sd(int e, int E, const int* __restrict__ ei,
                                        int& s, int& d) {
    if (e < E) { s = ei[e]; d = ei[E + e]; }
    else       { s = e - E; d = e - E; }      // appended self-loops
}

__device__ __forceinline__ float atomicMaxF(float* addr, float val) {
    // monotonic int/uint ordering trick; valid with -inf init
    if (val >= 0.0f)
        return __int_as_float(atomicMax((int*)addr, __float_as_int(val)));
    return __uint_as_float(atomicMin((unsigned int*)addr, __float_as_uint(val)));
}

// --------------- pass A: leaky-relu logit + segment max --------------------
__global__ void gat_edge_logits_max(const int* __restrict__ ei, int E, int Etot, int H,
                                    const float* __restrict__ alS,
                                    const float* __restrict__ alD,
                                    float* __restrict__ elg,
                                    float* __restrict__ mmax) {
    const int t = blockIdx.x * blockDim.x + threadIdx.x;
    if (t >= Etot * H) return;
    const int e = t / H, hh = t % H;
    int s, d;
    edge_sd(e, E, ei, s, d);
    float lg = alS[s * H + hh] + alD[d * H + hh];
    lg = lg > 0.0f ? lg : NEG_SLOPE * lg;
    elg[t] = lg;
    atomicMaxF(&mmax[d * H + hh], lg);
}

// --------------- pass B: exp + segment sum ---------------------------------
__global__ void gat_edge_exp_sum(const int* __restrict__ ei, int E, int Etot, int H,
                                 const float* __restrict__ mmax,
                                 float* __restrict__ elg,
                                 float* __restrict__ denom) {
    const int t = blockIdx.x * blockDim.x + threadIdx.x;
    if (t >= Etot * H) return;
    const int e = t / H, hh = t % H;
    int s, d;
    edge_sd(e, E, ei, s, d);
    const float ee = __expf(elg[t] - mmax[d * H + hh]);
    elg[t] = ee;
    atomicAdd(&denom[d * H + hh], ee);
}

// --------------- pass C: alpha-weighted message scatter --------------------
// one thread per (edge, channel); coalesced on channel; atomics land in L2.
__global__ void gat_edge_scatter(const int* __restrict__ ei, int E, int Etot,
                                 int H, int C, int ldH,
                                 const float* __restrict__ hin,
                                 const float* __restrict__ elg,
                                 const float* __restrict__ denom,
                                 float* __restrict__ aggout) {
    const int HC = H * C;
    const int t  = blockIdx.x * blockDim.x + threadIdx.x;
    if (t >= Etot * HC) return;
    const int e = t / HC, j = t % HC, hh = j / C;
    int s, d;
    edge_sd(e, E, ei, s, d);
    const float alpha = elg[e * H + hh] / (denom[d * H + hh] + 1e-16f);
    atomicAdd(&aggout[(long)d * HC + j], hin[(long)s * ldH + j] * alpha);
}

// --------------- epilogue: ReLU + BatchNorm (eval) -------------------------
__global__ void gat_relu_bn(const float* __restrict__ in, float* __restrict__ out,
                            const float* __restrict__ g, const float* __restrict__ b,
                            const float* __restrict__ m, const float* __restrict__ v,
                            int n, int HC) {
    const int i = blockIdx.x * blockDim.x + threadIdx.x;
    if (i >= n * HC) return;
    const int c = i % HC;
    float x = in[i];
    x = x > 0.0f ? x : 0.0f;
    out[i] = (x - m[c]) * rsqrtf(v[c] + BN_EPS) * g[c] + b[c];
}

// --------------- final log_softmax over D_OUT ------------------------------
__global__ void gat_log_softmax(const float* __restrict__ in, float* __restrict__ out,
                                int n, int d) {
    const int i = blockIdx.x * blockDim.x + threadIdx.x;
    if (i >= n) return;
    const float* row = in + (long)i * d;
    float mx = row[0];
    for (int j = 1; j < d; ++j) mx = fmaxf(mx, row[j]);
    float s = 0.0f;
    for (int j = 0; j < d; ++j) s += __expf(row[j] - mx);
    const float ls = __logf(s);
    for (int j = 0; j < d; ++j) out[(long)i * d + j] = row[j] - mx - ls;
}

// ---------------------------------------------------------------------------
extern "C" void kernel_launch(void* const* d_in, const int* in_sizes, int n_in,
                              void* d_out, int out_size, void* d_ws, size_t ws_size,
                              hipStream_t stream) {
    const float* x  = (const float*)d_in[0];
    const int*   ei = (const int*)d_in[1];
    const float* W1 = (const float*)d_in[2];
    const float* as1 = (const float*)d_in[3];
    const float* ad1 = (const float*)d_in[4];
    const float* b1  = (const float*)d_in[5];
    const float* W2 = (const float*)d_in[6];
    const float* as2 = (const float*)d_in[7];
    const float* ad2 = (const float*)d_in[8];
    const float* b2  = (const float*)d_in[9];
    const float* W3 = (const float*)d_in[10];
    const float* as3 = (const float*)d_in[11];
    const float* ad3 = (const float*)d_in[12];
    const float* b3  = (const float*)d_in[13];
    const float* bn1g = (const float*)d_in[14];
    const float* bn1b = (const float*)d_in[15];
    const float* bn1m = (const float*)d_in[16];
    const float* bn1v = (const float*)d_in[17];
    const float* bn2g = (const float*)d_in[18];
    const float* bn2b = (const float*)d_in[19];
    const float* bn2m = (const float*)d_in[20];
    const float* bn2v = (const float*)d_in[21];

    const int D_IN = 128, HID = 32, HEADS = 4, HC = HEADS * HID, D_OUT = 40;
    const int NPAD3 = 48;                         // D_OUT padded to 16-multiple
    const int n    = in_sizes[0] / D_IN;          // 50000 (multiple of 16)
    const int E    = in_sizes[1] / 2;             // 800000
    const int Etot = E + n;                       // + self loops

    // ---- carve workspace ----------------------------------------------------
    char*  ws  = (char*)d_ws;
    size_t off = 0;
    auto carve = [&](size_t bytes) -> float* {
        float* p = (float*)(ws + off);
        off += (bytes + 255) & ~(size_t)255;
        return p;
    };
    float* hproj  = carve((size_t)n * HC * sizeof(float));   // >= n*NPAD3 too
    float* aggout = carve((size_t)n * HC * sizeof(float));
    float* hcur   = carve((size_t)n * HC * sizeof(float));
    float* alS    = carve((size_t)n * HEADS * sizeof(float));
    float* alD    = carve((size_t)n * HEADS * sizeof(float));
    float* mmax   = carve((size_t)n * HEADS * sizeof(float));
    float* denom  = carve((size_t)n * HEADS * sizeof(float));
    float* elg    = carve((size_t)Etot * HEADS * sizeof(float));
    float* wpad   = carve((size_t)KDIM * NPAD3 * sizeof(float));
    (void)ws_size; (void)n_in; (void)out_size;

    const int TB = 256;
    auto blocks = [](long work, int tb) { return (int)((work + tb - 1) / tb); };

    const int mtiles     = n >> 4;
    const int gemm_grid  = blocks((long)mtiles * 32, TB);

    // ---- shared post-GEMM stages of one GAT layer ---------------------------
    auto attention = [&](int H, int C, int ldH,
                         const float* as, const float* ad, const float* bias) {
        const int hc = H * C;
        gat_al_kernel<<<blocks((long)n * H, TB), TB, 0, stream>>>(
            hproj, ldH, as, ad, alS, alD, n, H, C);
        gat_init_kernel<<<blocks((long)n * hc, TB), TB, 0, stream>>>(
            aggout, mmax, denom, bias, n, hc, H);
        gat_edge_logits_max<<<blocks((long)Etot * H, TB), TB, 0, stream>>>(
            ei, E, Etot, H, alS, alD, elg, mmax);
        gat_edge_exp_sum<<<blocks((long)Etot * H, TB), TB, 0, stream>>>(
            ei, E, Etot, H, mmax, elg, denom);
        gat_edge_scatter<<<blocks((long)Etot * hc, TB), TB, 0, stream>>>(
            ei, E, Etot, H, C, ldH, hproj, elg, denom, aggout);
    };

    // Layer 1: x[128] -> 4 heads x 32 (ldH = 128), then ReLU + BN1 -> hcur
    gat_gemm_wmma_f32<8><<<gemm_grid, TB, 0, stream>>>(x, W1, hproj, n, HC);
    attention(HEADS, HID, HC, as1, ad1, b1);
    gat_relu_bn<<<blocks((long)n * HC, TB), TB, 0, stream>>>(
        aggout, hcur, bn1g, bn1b, bn1m, bn1v, n, HC);

    // Layer 2: hcur[128] -> 4 heads x 32 (ldH = 128), then ReLU + BN2 -> hcur
    gat_gemm_wmma_f32<8><<<gemm_grid, TB, 0, stream>>>(hcur, W2, hproj, n, HC);
    attention(HEADS, HID, HC, as2, ad2, b2);
    gat_relu_bn<<<blocks((long)n * HC, TB), TB, 0, stream>>>(
        aggout, hcur, bn2g, bn2b, bn2m, bn2v, n, HC);

    // Layer 3: hcur[128] -> 1 head x 40 via zero-padded W3 (ldH = 48),
    // then log_softmax -> d_out
    gat_pad_weight<<<blocks((long)KDIM * NPAD3, TB), TB, 0, stream>>>(
        W3, wpad, KDIM, D_OUT, NPAD3);
    gat_gemm_wmma_f32<3><<<gemm_grid, TB, 0, stream>>>(hcur, wpad, hproj, n, NPAD3);
    attention(1, D_OUT, NPAD3, as3, ad3, b3);
    gat_log_softmax<<<blocks((long)n, TB), TB, 0, stream>>>(
        aggout, (float*)d_out, n, D_OUT);
}